// RPNInference_56667798503872
// MI455X (gfx1250) — compile-verified
//
#include <hip/hip_runtime.h>
#include <hip/hip_bf16.h>
#include <math.h>

// ---------------- problem constants (from reference) ----------------
#define NN   16
#define AA   3
#define HH   256
#define WW   256
#define KK   (AA*HH*WW)        // 196608 anchors per image
#define PRE  2000
#define POST 1000
#define NMS_T 0.7f
#define NEGV (-1e30f)
#define NEGHALF (-5e29f)
#define BBOX_CLIP 4.135166556742356f   // log(1000/16)
#define IMGW 1024.0f
#define IMGH 1024.0f
#define MS   8                 // meta stride (ints per image)
// meta layout: [0]=b1 [1]=t1 [2]=b2 [3]=t2 [4]=thrKey [5]=numGt

typedef __attribute__((ext_vector_type(4))) unsigned int u32x4;
typedef __attribute__((ext_vector_type(8))) int          i32x8;
typedef __attribute__((ext_vector_type(4))) int          i32x4;
typedef __attribute__((ext_vector_type(2))) float        v2f;
typedef __attribute__((ext_vector_type(8))) float        v8f;

// ---------------- CDNA5 feature availability (device pass only) ----------------
#if defined(__HIP_DEVICE_COMPILE__) && defined(__has_builtin)
#  if __has_builtin(__builtin_amdgcn_tensor_load_to_lds) && __has_builtin(__builtin_amdgcn_s_wait_tensorcnt)
#    define TDM_OK 1
#  endif
#  if __has_builtin(__builtin_amdgcn_wmma_f32_16x16x4_f32)
#    define WMMA_OK 1
#  endif
#endif
#ifndef TDM_OK
#define TDM_OK 0
#endif
#ifndef WMMA_OK
#define WMMA_OK 0
#endif

#if defined(__HIP_DEVICE_COMPILE__)
#  if TDM_OK
#    warning "gfx1250 device pass: TDM tensor_load_to_lds path ENABLED"
#  else
#    warning "gfx1250 device pass: TDM builtin NOT available - plain-load fallback"
#  endif
#  if WMMA_OK
#    warning "gfx1250 device pass: WMMA f32_16x16x4 scan ENABLED"
#  else
#    warning "gfx1250 device pass: WMMA f32_16x16x4 builtin NOT available - serial scan fallback"
#  endif
#endif

#if TDM_OK
// 1-D contiguous TDM copy global -> LDS, nbytes % 8 == 0.
__device__ __forceinline__ void tdm_load_1d(void* ldsPtr, const void* gPtr, unsigned nbytes) {
  unsigned long long ga = (unsigned long long)gPtr;
  unsigned lds = (unsigned)(unsigned long long)ldsPtr;   // low 32 bits = LDS offset
  unsigned ne = nbytes >> 3;                             // 8-byte elements
  u32x4 g0;
  g0[0] = 1u;                                            // count=1, user descriptor
  g0[1] = lds;                                           // lds_addr
  g0[2] = (unsigned)(ga & 0xFFFFFFFFu);                  // global_addr[31:0]
  g0[3] = (unsigned)((ga >> 32) & 0x01FFFFFFu) | (2u << 30); // addr[56:32] | type=2
  i32x8 g1;
  g1[0] = (int)(3u << 16);                               // wg_mask=0, data_size=3 (8B)
  g1[1] = (int)((ne & 0xFFFFu) << 16);                   // tensor_dim0[15:0]
  g1[2] = (int)((ne >> 16) | (1u << 16));                // tensor_dim0[31:16] | tensor_dim1=1 lo
  g1[3] = (int)((ne & 0xFFFFu) << 16);                   // tensor_dim1 hi=0 | tile_dim0=ne
  g1[4] = 1;                                             // tile_dim1=1 | tile_dim2=0
  g1[5] = (int)ne;                                       // tensor_dim0_stride lo32
  g1[6] = 0;
  g1[7] = 0;
  i32x4 g2; g2[0] = 1; g2[1] = 1; g2[2] = 0; g2[3] = 0;  // tensor_dim2=1, dim3=1, tile_dim3=0
  i32x4 g3; g3[0] = 0; g3[1] = (int)(1u << 16); g3[2] = 0; g3[3] = 0; // tensor_dim4=1, tile_dim4=0
#if __has_include(<hip/amd_detail/amd_gfx1250_TDM.h>)
  i32x8 gz = {0,0,0,0,0,0,0,0};
  __builtin_amdgcn_tensor_load_to_lds(g0, g1, g2, g3, gz, 0);   // clang-23 6-arg form
#else
  __builtin_amdgcn_tensor_load_to_lds(g0, g1, g2, g3, 0);       // ROCm 7.2 5-arg form
#endif
}
#endif

__device__ __forceinline__ float sigm(float x) { return 1.0f / (1.0f + expf(-x)); }

// ---------------- WMMA-based suffix-sum scan helpers ----------------
// Computes sP[i] = sum_{j<=i} rev[j], rev[j] = h[nbins-1-j]  (inclusive prefix of
// the top-down order). Exact: all counts < 2^24 so f32 accumulation is integral.
__device__ __forceinline__ void wavePrefixRev(const unsigned* __restrict__ h, int nbins,
                                              float* __restrict__ sP) {
#if WMMA_OK
  const int lane = (int)(threadIdx.x & 31);
  const int nchunks = nbins >> 8;                 // 256 bins per chunk
  const int m   = lane & 15;                      // A-matrix row / B col / D col
  const int khi = (lane < 16) ? 0 : 2;            // K sub-index pair per lane half
  for (int c = 0; c < nchunks; ++c) {
    const int base = c << 8;
    v8f acc = {0.0f,0.0f,0.0f,0.0f,0.0f,0.0f,0.0f,0.0f};
    #pragma unroll
    for (int kc = 0; kc < 4; ++kc) {              // Y = L(16x16) @ X(16x16), K chunks of 4
      const int k0 = 4*kc + khi;
      v2f a, b;
      a.x = (k0     <= m) ? 1.0f : 0.0f;          // L[M][k0]   (inclusive lower-tri ones)
      a.y = (k0 + 1 <= m) ? 1.0f : 0.0f;          // L[M][k0+1]
      b.x = (float)h[nbins - 1 - (base + m*16 + k0)];     // X[k0  ][n=m]
      b.y = (float)h[nbins - 1 - (base + m*16 + k0 + 1)]; // X[k0+1][n=m]
      acc = __builtin_amdgcn_wmma_f32_16x16x4_f32(false, a, false, b, (short)0, acc,
                                                  false, false);
    }
    const int rowBase = (lane < 16) ? 0 : 8;      // D: VGPR j holds rows j / j+8
    #pragma unroll
    for (int j = 0; j < 8; ++j)
      sP[base + m*16 + rowBase + j] = acc[j];
  }
  __syncthreads();
  if (lane == 0) {                                // carry across 16-wide segments (exact)
    float carry = 0.0f;
    const int nseg = nbins >> 4;
    for (int s = 0; s < nseg; ++s) {
      float tot = sP[s*16 + 15];
      if (carry != 0.0f)
        for (int i = 0; i < 16; ++i) sP[s*16 + i] += carry;
      carry += tot;
    }
  }
  __syncthreads();
#else
  (void)h; (void)nbins; (void)sP;
#endif
}

__device__ __forceinline__ int findThreshF(const float* __restrict__ sP, int nbins,
                                           int target, int* cumBefore) {
  const float ft = (float)target;
  int i = 0;
  while (i < nbins && sP[i] < ft) ++i;
  *cumBefore = (i > 0) ? (int)sP[i - 1] : 0;
  return i;
}

// serial fallback (also used when WMMA builtin unavailable)
__device__ __forceinline__ void serialScan(const unsigned* __restrict__ h, int nbins,
                                           int target, int* binOut, int* remOut) {
  unsigned cum = 0; int bb = 0;
  for (int b = nbins - 1; b >= 0; --b) {
    unsigned c = h[b];
    if ((int)(cum + c) >= target) { bb = b; break; }
    cum += c;
  }
  *binOut = bb; *remOut = target - (int)cum;
}

// ---------------- kernels ----------------
__global__ void kZero(unsigned* h1, unsigned* h2, unsigned* h3, int* cnts,
                      float* selBoxes, float* selScores) {
  int t = blockIdx.x * blockDim.x + threadIdx.x;
  int st = gridDim.x * blockDim.x;
  for (int i = t; i < NN*4096; i += st) h1[i] = 0u;
  for (int i = t; i < NN*4096; i += st) h2[i] = 0u;
  for (int i = t; i < NN*256;  i += st) h3[i] = 0u;
  for (int i = t; i < NN*2;    i += st) cnts[i] = 0;
  for (int i = t; i < NN*PRE;  i += st) selScores[i] = NEGV;
  for (int i = t; i < NN*PRE*4;i += st) selBoxes[i] = 0.0f;
}

__global__ void kHist1(const float* __restrict__ obj, unsigned* __restrict__ h1) {
  long long t = (long long)blockIdx.x * blockDim.x + threadIdx.x;
  const long long total = (long long)NN * KK;
  if (t >= total) return;
  if (t + 8192 < total) __builtin_prefetch(obj + t + 8192, 0, 0);  // global_prefetch_b8
  int n = (int)(t / KK);
  unsigned key = __float_as_uint(sigm(obj[t]));
  atomicAdd(&h1[n*4096 + (key >> 20)], 1u);
}

__global__ void kScanW1(const unsigned* __restrict__ h1, int* __restrict__ meta) {
  __shared__ float sP[4096];
  const int n = blockIdx.x;
  const unsigned* h = h1 + n*4096;
#if WMMA_OK
  wavePrefixRev(h, 4096, sP);
  if (threadIdx.x == 0) {
    int cb; int i = findThreshF(sP, 4096, PRE, &cb);
    meta[n*MS + 0] = 4095 - i;
    meta[n*MS + 1] = PRE - cb;
  }
#else
  if (threadIdx.x == 0) {
    int b, r; serialScan(h, 4096, PRE, &b, &r);
    meta[n*MS + 0] = b; meta[n*MS + 1] = r;
  }
  (void)sP;
#endif
}

__global__ void kHist2(const float* __restrict__ obj, const int* __restrict__ meta,
                       unsigned* __restrict__ h2) {
  long long t = (long long)blockIdx.x * blockDim.x + threadIdx.x;
  if (t >= (long long)NN * KK) return;
  int n = (int)(t / KK);
  unsigned key = __float_as_uint(sigm(obj[t]));
  if ((int)(key >> 20) == meta[n*MS + 0])
    atomicAdd(&h2[n*4096 + ((key >> 8) & 0xFFFu)], 1u);
}

__global__ void kScanW2(const unsigned* __restrict__ h2, int* __restrict__ meta) {
  __shared__ float sP[4096];
  const int n = blockIdx.x;
  const unsigned* h = h2 + n*4096;
  const int target = meta[n*MS + 1];
#if WMMA_OK
  wavePrefixRev(h, 4096, sP);
  if (threadIdx.x == 0) {
    int cb; int i = findThreshF(sP, 4096, target, &cb);
    meta[n*MS + 2] = 4095 - i;
    meta[n*MS + 3] = target - cb;
  }
#else
  if (threadIdx.x == 0) {
    int b, r; serialScan(h, 4096, target, &b, &r);
    meta[n*MS + 2] = b; meta[n*MS + 3] = r;
  }
  (void)sP;
#endif
}

__global__ void kHist3(const float* __restrict__ obj, const int* __restrict__ meta,
                       unsigned* __restrict__ h3) {
  long long t = (long long)blockIdx.x * blockDim.x + threadIdx.x;
  if (t >= (long long)NN * KK) return;
  int n = (int)(t / KK);
  unsigned key = __float_as_uint(sigm(obj[t]));
  unsigned hi = ((unsigned)meta[n*MS + 0] << 12) | (unsigned)meta[n*MS + 2];
  if ((key >> 8) == hi) atomicAdd(&h3[n*256 + (key & 0xFFu)], 1u);
}

__global__ void kScanW3(const unsigned* __restrict__ h3, int* __restrict__ meta) {
  __shared__ float sP[256];
  const int n = blockIdx.x;
  const unsigned* h = h3 + n*256;
  const int target = meta[n*MS + 3];
#if WMMA_OK
  wavePrefixRev(h, 256, sP);
  if (threadIdx.x == 0) {
    int cb; int i = findThreshF(sP, 256, target, &cb);
    int b3 = 255 - i;
    int need3 = target - cb;
    unsigned thr = ((unsigned)meta[n*MS + 0] << 20) | ((unsigned)meta[n*MS + 2] << 8) |
                   (unsigned)b3;
    meta[n*MS + 4] = (int)thr;
    meta[n*MS + 5] = PRE - need3;
  }
#else
  if (threadIdx.x == 0) {
    int b3, need3; serialScan(h, 256, target, &b3, &need3);
    unsigned thr = ((unsigned)meta[n*MS + 0] << 20) | ((unsigned)meta[n*MS + 2] << 8) |
                   (unsigned)b3;
    meta[n*MS + 4] = (int)thr;
    meta[n*MS + 5] = PRE - need3;
  }
  (void)sP;
#endif
}

// select the exact top-2000 per image, decode+clip+min-size only for winners
__global__ void kCompact(const float* __restrict__ obj, const float* __restrict__ br,
                         const float* __restrict__ anch, const int* __restrict__ meta,
                         int* __restrict__ cnts,
                         float* __restrict__ selBoxes, float* __restrict__ selScores) {
  long long t = (long long)blockIdx.x * blockDim.x + threadIdx.x;
  if (t >= (long long)NN * KK) return;
  int n = (int)(t / KK);
  int r = (int)(t % KK);
  int a = r / (HH*WW);
  int hw = r % (HH*WW);
  int h = hw / WW, w = hw % WW;

  float sc = sigm(obj[t]);
  unsigned key = __float_as_uint(sc);
  unsigned thr = (unsigned)meta[n*MS + 4];
  int slot = -1;
  if (key > thr) {
    slot = atomicAdd(&cnts[n*2 + 0], 1);
  } else if (key == thr) {
    int numGt = meta[n*MS + 5];
    int e = atomicAdd(&cnts[n*2 + 1], 1);
    if (e < PRE - numGt) slot = numGt + e;
  }
  if (slot < 0) return;

  int k = hw * AA + a;                       // anchor index in (h,w,a) order
  const float* an = anch + ((size_t)n*KK + k)*4;
  float ax1 = an[0], ay1 = an[1], ax2 = an[2], ay2 = an[3];
  float aw = ax2 - ax1 + 1.0f, ah = ay2 - ay1 + 1.0f;
  float cx = ax1 + 0.5f*aw,    cy = ay1 + 0.5f*ah;
  const size_t plane = (size_t)HH*WW;
  size_t bb = (((size_t)n*(AA*4) + a*4)*HH + h)*WW + w;
  float dx = br[bb], dy = br[bb + plane];
  float dw = fminf(br[bb + 2*plane], BBOX_CLIP);
  float dh = fminf(br[bb + 3*plane], BBOX_CLIP);
  float pcx = dx*aw + cx, pcy = dy*ah + cy;
  float pw = expf(dw)*aw,  ph = expf(dh)*ah;
  float x1 = pcx - 0.5f*pw, y1 = pcy - 0.5f*ph;
  float x2 = pcx + 0.5f*pw - 1.0f, y2 = pcy + 0.5f*ph - 1.0f;
  x1 = fminf(fmaxf(x1, 0.0f), IMGW - 1.0f);
  y1 = fminf(fmaxf(y1, 0.0f), IMGH - 1.0f);
  x2 = fminf(fmaxf(x2, 0.0f), IMGW - 1.0f);
  y2 = fminf(fmaxf(y2, 0.0f), IMGH - 1.0f);
  bool keep = (x2 - x1 + 1.0f >= 0.0f) && (y2 - y1 + 1.0f >= 0.0f);

  float* sb = selBoxes + ((size_t)n*PRE + slot)*4;
  sb[0] = x1; sb[1] = y1; sb[2] = x2; sb[3] = y2;
  selScores[(size_t)n*PRE + slot] = keep ? sc : NEGV;
}

// one workgroup per image; whole working set in LDS; TDM-staged
__global__ __launch_bounds__(1024)
void kNMS(const float* __restrict__ selBoxes, const float* __restrict__ selScores,
          float* __restrict__ outBoxes, float* __restrict__ outScores) {
  __shared__ float4 sBox[PRE];     // 32000 B
  __shared__ float  sOrig[PRE];    //  8000 B
  __shared__ float  sWork[PRE];    //  8000 B
  __shared__ float  rSc[32];
  __shared__ int    rIx[32];
  __shared__ float  sBestSc;
  __shared__ int    sBestIx;

  const int n = blockIdx.x;
  const int tid = threadIdx.x;
  const int lane = tid & 31, wv = tid >> 5;
  const float* gBox = selBoxes + (size_t)n*PRE*4;
  const float* gSc  = selScores + (size_t)n*PRE;

#if TDM_OK
  if (tid < 32) {
    tdm_load_1d(&sBox[0],  gBox, PRE*4*sizeof(float));
    tdm_load_1d(&sOrig[0], gSc,  PRE*sizeof(float));
    __builtin_amdgcn_s_wait_tensorcnt(0);
  }
  __syncthreads();
  for (int e = tid; e < PRE; e += 1024) sWork[e] = sOrig[e];
#else
  for (int e = tid; e < PRE; e += 1024) {
    sBox[e] = ((const float4*)gBox)[e];
    float s = gSc[e];
    sOrig[e] = s; sWork[e] = s;
  }
#endif
  __syncthreads();

  for (int it = 0; it < POST; ++it) {
    // ---- parallel argmax (first-index tie-break) ----
    float b = -3e30f; int bi = 0;
    float v0 = sWork[tid];
    if (v0 > b) { b = v0; bi = tid; }
    int e1 = tid + 1024;
    if (e1 < PRE) { float v1 = sWork[e1]; if (v1 > b) { b = v1; bi = e1; } }
    for (int off = 16; off; off >>= 1) {
      float ob = __shfl_down(b, off, 32);
      int   oi = __shfl_down(bi, off, 32);
      if (ob > b || (ob == b && oi < bi)) { b = ob; bi = oi; }
    }
    if (lane == 0) { rSc[wv] = b; rIx[wv] = bi; }
    __syncthreads();
    if (wv == 0) {
      b = rSc[lane]; bi = rIx[lane];
      for (int off = 16; off; off >>= 1) {
        float ob = __shfl_down(b, off, 32);
        int   oi = __shfl_down(bi, off, 32);
        if (ob > b || (ob == b && oi < bi)) { b = ob; bi = oi; }
      }
      if (lane == 0) { sBestSc = b; sBestIx = bi; }
    }
    __syncthreads();

    const int best = sBestIx;
    const bool valid = sBestSc > NEGHALF;
    const float4 bbx = sBox[best];
    if (tid == 0) {
      float* op = outBoxes + ((size_t)n*POST + it)*4;
      op[0] = valid ? bbx.x : 0.0f;
      op[1] = valid ? bbx.y : 0.0f;
      op[2] = valid ? bbx.z : 0.0f;
      op[3] = valid ? bbx.w : 0.0f;
      outScores[(size_t)n*POST + it] = valid ? sOrig[best] : NEGV;
    }
    // ---- suppress by IoU against chosen box (matches ref: done even if !valid) ----
    const float areaB = (bbx.z - bbx.x + 1.0f) * (bbx.w - bbx.y + 1.0f);
    for (int e = tid; e < PRE; e += 1024) {
      float4 c = sBox[e];
      float iw = fminf(c.z, bbx.z) - fmaxf(c.x, bbx.x) + 1.0f; iw = fmaxf(iw, 0.0f);
      float ih = fminf(c.w, bbx.w) - fmaxf(c.y, bbx.y) + 1.0f; ih = fmaxf(ih, 0.0f);
      float inter = iw * ih;
      float areaC = (c.z - c.x + 1.0f) * (c.w - c.y + 1.0f);
      float iou = inter / (areaC + areaB - inter);
      if (iou > NMS_T) sWork[e] = NEGV;
    }
    __syncthreads();
  }
}

// ---------------- host launcher ----------------
extern "C" void kernel_launch(void* const* d_in, const int* in_sizes, int n_in,
                              void* d_out, int out_size, void* d_ws, size_t ws_size,
                              hipStream_t stream) {
  const float* anchors = (const float*)d_in[0];
  const float* objectness = (const float*)d_in[1];
  const float* box_regression = (const float*)d_in[2];
  float* out = (float*)d_out;
  float* outBoxes  = out;                 // N*POST*4 floats
  float* outScores = out + NN*POST*4;     // N*POST floats

  size_t off = 0;
  auto alloc = [&](size_t bytes) { size_t o = off; off = (off + bytes + 255) & ~(size_t)255; return o; };
  char* ws = (char*)d_ws;
  unsigned* h1   = (unsigned*)(ws + alloc((size_t)NN*4096*4));
  unsigned* h2   = (unsigned*)(ws + alloc((size_t)NN*4096*4));
  unsigned* h3   = (unsigned*)(ws + alloc((size_t)NN*256*4));
  int*      meta = (int*)     (ws + alloc((size_t)NN*MS*4));
  int*      cnts = (int*)     (ws + alloc((size_t)NN*2*4));
  float* selBoxes  = (float*) (ws + alloc((size_t)NN*PRE*4*4));
  float* selScores = (float*) (ws + alloc((size_t)NN*PRE*4));
  (void)ws_size; (void)n_in; (void)in_sizes; (void)out_size;

  const long long total = (long long)NN * KK;   // 3,145,728
  const int tpb = 256;
  const int blocks = (int)((total + tpb - 1) / tpb);

  kZero<<<512, 256, 0, stream>>>(h1, h2, h3, cnts, selBoxes, selScores);
  kHist1<<<blocks, tpb, 0, stream>>>(objectness, h1);
  kScanW1<<<NN, 32, 0, stream>>>(h1, meta);
  kHist2<<<blocks, tpb, 0, stream>>>(objectness, meta, h2);
  kScanW2<<<NN, 32, 0, stream>>>(h2, meta);
  kHist3<<<blocks, tpb, 0, stream>>>(objectness, meta, h3);
  kScanW3<<<NN, 32, 0, stream>>>(h3, meta);
  kCompact<<<blocks, tpb, 0, stream>>>(objectness, box_regression, anchors, meta, cnts,
                                       selBoxes, selScores);
  kNMS<<<NN, 1024, 0, stream>>>(selBoxes, selScores, outBoxes, outScores);
}